// HyperboloidEmbeddingLayerGaussian_24086176596781
// MI455X (gfx1250) — compile-verified
//
#include <hip/hip_runtime.h>
#include <math.h>

#define EPSF 1e-7f
#define DDIM 128
#define EDIM 129
#define NS_  10
#define NIDX 11
#define WPB  8          // waves per block (256 threads)
#define ROWPAD 132      // padded LDS row stride (floats): 528B, odd bank phase per row
#define WROWS 14        // 10 tgt + src + r + ones + zero
#define WSIZE (WROWS * ROWPAD)

typedef __attribute__((ext_vector_type(2))) float v2f;
typedef __attribute__((ext_vector_type(8))) float v8f;

__device__ __forceinline__ float sig_xform(float s) {
  // where(s>0, s, (1-EPS)*expm1(min(s,0))) + 1, clamped at EPS
  float neg = (1.0f - EPSF) * expm1f(fminf(s, 0.0f));
  float v = (s > 0.0f) ? s : neg;
  return fmaxf(v + 1.0f, EPSF);
}

// Extract D[M][N] from 16x16 f32 C/D tile: VGPR v holds M=v (lanes0-15) / M=v+8 (lanes16-31), N=lane&15
template<int M, int N>
__device__ __forceinline__ float dget(v8f c) {
  constexpr int v = M & 7;
  constexpr int srcLane = ((M >> 3) << 4) | N;
  float x = c[v];
  return __shfl(x, srcLane, 32);
}

__device__ __forceinline__ float wave_sum(float x) {
#pragma unroll
  for (int off = 16; off >= 1; off >>= 1)
    x += __shfl_xor(x, off, 32);
  return x;
}

template<int S>
__device__ __forceinline__ void finalize(v8f acc, float Q, float SS, float srcD,
                                         float tgtD_lane, const float* tracePart,
                                         const float* Lpart, float L0,
                                         float* __restrict__ out, long long b, int lane) {
  float P  = dget<S, 0>(acc);       // sum src*tgt over spatial dims
  float ST = dget<S, 1>(acc);       // sum src*tgt*r
  float T2 = dget<S, 2 + S>(acc);   // sum tgt*tgt*r
  float trace = wave_sum(tracePart[S]);
  float Ls    = wave_sum(Lpart[S]);
  float tDs = __shfl(tgtD_lane, S, 32);     // tgt time component

  float a_raw = srcD * tDs - P;                       // alpha = -mdot(src,tgt)
  float alpha = 1.0f + fmaxf(a_raw - 1.0f, EPSF);
  float denom = fmaxf(sqrtf(fmaxf(alpha * alpha - 1.0f, 0.0f)), EPSF);
  float k = acoshf(alpha) / denom;                    // log_map scale
  float S1 = k * (P - alpha * Q);                     // sum src_i * logmap_i
  float lmD = k * (tDs - alpha * srcD);               // logmap time component
  float a2 = srcD;                                    // -mdot(src, mu0)
  float cc = (-a2 * S1 - (1.0f - a2 * srcD) * lmD) / fmaxf(a2 + 1.0f, EPSF);
  // uu = sum (logmap_i + cc*src_i)^2 * r_i, expanded
  float uu = k * k * (T2 - 2.0f * alpha * ST + alpha * alpha * SS)
           + 2.0f * cc * k * (ST - alpha * SS)
           + cc * cc * SS;
  float res = 0.5f * (trace + uu - (float)DDIM - (Ls - L0));
  if (lane == 0) out[b * NS_ + S] = res;
}

__global__ __launch_bounds__(256)
void hyperboloid_gaussian_kernel(const int* __restrict__ idx,
                                 const float* __restrict__ emb,
                                 const float* __restrict__ cov,
                                 float* __restrict__ out, int Btot) {
  __shared__ float lds[WPB * WSIZE];
  const int lane = threadIdx.x & 31;
  const int wave = threadIdx.x >> 5;
  const long long b = (long long)blockIdx.x * WPB + wave;
  if (b >= Btot) return;                 // wave-uniform

  float* base  = lds + wave * WSIZE;
  float* tgtL  = base;                   // rows 0..9
  float* srcL  = base + 10 * ROWPAD;
  float* rL    = base + 11 * ROWPAD;
  float* onesL = base + 12 * ROWPAD;
  float* zeroL = base + 13 * ROWPAD;

  // ---- constant rows ----
#pragma unroll
  for (int j = 0; j < 4; ++j) {
    int d = lane + 32 * j;
    onesL[d] = 1.0f;
    zeroL[d] = 0.0f;
  }

  // ---- gather indices (lane t holds idx[b][t] for t<11) ----
  int idxv = idx[b * NIDX + ((lane < NIDX) ? lane : 0)];
  int srcIdx = __shfl(idxv, 0, 32);

  // ---- source embedding spatial dims -> LDS (stride-32 coalesced b32) ----
  const float* srcRow = emb + (long long)srcIdx * EDIM;
#pragma unroll
  for (int j = 0; j < 4; ++j) {
    int d = lane + 32 * j;
    srcL[d] = srcRow[d];
  }
  float srcD = srcRow[DDIM];

  // ---- source covariance -> sigma transform -> r = 1/ssig (float4, 16B aligned) ----
  float4 sc = reinterpret_cast<const float4*>(cov + (long long)srcIdx * DDIM)[lane];
  float s0 = sig_xform(sc.x), s1 = sig_xform(sc.y), s2 = sig_xform(sc.z), s3 = sig_xform(sc.w);
  float4 rr = make_float4(1.0f / s0, 1.0f / s1, 1.0f / s2, 1.0f / s3);
  float L0 = wave_sum(logf(s0) + logf(s1) + logf(s2) + logf(s3));
  *reinterpret_cast<float4*>(rL + 4 * lane) = rr;

  // ---- targets: embeddings -> LDS, covariance -> trace / logdet partials ----
  float tracePart[NS_], Lpart[NS_];
#pragma unroll
  for (int s = 0; s < NS_; ++s) {
    int tIdx = __shfl(idxv, 1 + s, 32);
    const float* tRow = emb + (long long)tIdx * EDIM;
#pragma unroll
    for (int j = 0; j < 4; ++j) {
      int d = lane + 32 * j;
      tgtL[s * ROWPAD + d] = tRow[d];
    }
    float4 tc = reinterpret_cast<const float4*>(cov + (long long)tIdx * DDIM)[lane];
    float t0 = sig_xform(tc.x), t1 = sig_xform(tc.y), t2 = sig_xform(tc.z), t3 = sig_xform(tc.w);
    tracePart[s] = t0 * rr.x + t1 * rr.y + t2 * rr.z + t3 * rr.w;
    Lpart[s] = logf(t0) + logf(t1) + logf(t2) + logf(t3);
  }
  // target time components (lane t<10 keeps tgt_t[128])
  int tIdxL = __shfl(idxv, (lane < NS_) ? (1 + lane) : 1, 32);
  float tgtD_lane = emb[(long long)tIdxL * EDIM + DDIM];

  // ---- WMMA f32 16x16x4, chained over 32 K-chunks of 4 dims (branchless) ----
  // A rows 0..9 = tgt_s, row 10 = src, rows 11..15 = 0.
  // B col0 = src, col1 = src*r, col(2+s) = tgt_s*r, cols 12..15 = 0.
  const int m  = lane & 15;   // A row / B col handled by this lane
  const int hi = lane >> 4;   // K sub-pair select: vgpr0/1 hold K = 2*hi, 2*hi+1

  const float* Arow  = (m < NS_) ? (tgtL + m * ROWPAD) : ((m == 10) ? srcL : zeroL);
  const float* Bval  = (m < 2) ? srcL : ((m < 12) ? (tgtL + (m - 2) * ROWPAD) : zeroL);
  const float* Bscal = (m >= 1 && m < 12) ? rL : onesL;
  Arow  += 2 * hi;
  Bval  += 2 * hi;
  Bscal += 2 * hi;

  v8f acc = {0.f, 0.f, 0.f, 0.f, 0.f, 0.f, 0.f, 0.f};
#pragma unroll
  for (int kb = 0; kb < DDIM; kb += 4) {
    v2f a  = *reinterpret_cast<const v2f*>(Arow  + kb);
    v2f bv = *reinterpret_cast<const v2f*>(Bval  + kb);
    v2f bs = *reinterpret_cast<const v2f*>(Bscal + kb);
    v2f bf = bv * bs;
    acc = __builtin_amdgcn_wmma_f32_16x16x4_f32(false, a, false, bf, (short)0, acc,
                                                false, false);
  }

  float Q  = dget<10, 0>(acc);   // sum src^2
  float SS = dget<10, 1>(acc);   // sum src^2 * r

  finalize<0>(acc, Q, SS, srcD, tgtD_lane, tracePart, Lpart, L0, out, b, lane);
  finalize<1>(acc, Q, SS, srcD, tgtD_lane, tracePart, Lpart, L0, out, b, lane);
  finalize<2>(acc, Q, SS, srcD, tgtD_lane, tracePart, Lpart, L0, out, b, lane);
  finalize<3>(acc, Q, SS, srcD, tgtD_lane, tracePart, Lpart, L0, out, b, lane);
  finalize<4>(acc, Q, SS, srcD, tgtD_lane, tracePart, Lpart, L0, out, b, lane);
  finalize<5>(acc, Q, SS, srcD, tgtD_lane, tracePart, Lpart, L0, out, b, lane);
  finalize<6>(acc, Q, SS, srcD, tgtD_lane, tracePart, Lpart, L0, out, b, lane);
  finalize<7>(acc, Q, SS, srcD, tgtD_lane, tracePart, Lpart, L0, out, b, lane);
  finalize<8>(acc, Q, SS, srcD, tgtD_lane, tracePart, Lpart, L0, out, b, lane);
  finalize<9>(acc, Q, SS, srcD, tgtD_lane, tracePart, Lpart, L0, out, b, lane);
}

extern "C" void kernel_launch(void* const* d_in, const int* in_sizes, int n_in,
                              void* d_out, int out_size, void* d_ws, size_t ws_size,
                              hipStream_t stream) {
  const int*   idx = (const int*)d_in[0];
  const float* emb = (const float*)d_in[1];
  const float* cov = (const float*)d_in[2];
  float* out = (float*)d_out;
  const int B = in_sizes[0] / NIDX;     // idx is (B, 11)
  const int grid = (B + WPB - 1) / WPB;
  hipLaunchKernelGGL(hyperboloid_gaussian_kernel, dim3(grid), dim3(256), 0, stream,
                     idx, emb, cov, out, B);
}